// GraphFilterBatchAttentional_22677427322942
// MI455X (gfx1250) — compile-verified
//
#include <hip/hip_runtime.h>
#include <hip/hip_bf16.h>
#include <math.h>

// Problem constants (match reference)
#define BB 8
#define GG 256
#define NN 1024
#define PP 4
#define FF 256
#define KCH 4
#define ATT_SLOPE 0.2f
#define OUT_SLOPE 0.01f   // jax.nn.leaky_relu default

#define KBLK 32           // B-panel depth staged in LDS per iteration

typedef __attribute__((ext_vector_type(2))) float v2f;
typedef __attribute__((ext_vector_type(8))) float v8f;

// ---------------------------------------------------------------------------
// Kernel 1: fold mixer into weight:  v1[p,g] = sum_f a1[p,f]*W[p,f,g]
// ---------------------------------------------------------------------------
__global__ __launch_bounds__(256) void precompute_v(
    const float* __restrict__ mixer,   // (P, 2F)
    const float* __restrict__ weight,  // (P, F, G)
    const float* __restrict__ wbias,   // (P, F)
    float* __restrict__ v1, float* __restrict__ v2, float* __restrict__ bb)
{
    int idx = blockIdx.x * blockDim.x + threadIdx.x;   // P*G threads
    if (idx >= PP * GG) return;
    int p = idx / GG, g = idx % GG;
    const float* mx = mixer + p * (2 * FF);
    const float* w  = weight + (long)p * FF * GG + g;
    float s1 = 0.f, s2 = 0.f;
    for (int f = 0; f < FF; ++f) {
        float wv = w[(long)f * GG];
        s1 += mx[f] * wv;
        s2 += mx[FF + f] * wv;
    }
    v1[idx] = s1;
    v2[idx] = s2;
    if (g == 0) {
        const float* wb = wbias + p * FF;
        float c1 = 0.f, c2 = 0.f;
        for (int f = 0; f < FF; ++f) { c1 += mx[f] * wb[f]; c2 += mx[FF + f] * wb[f]; }
        bb[p] = c1; bb[PP + p] = c2;
    }
}

// ---------------------------------------------------------------------------
// Kernel 2: e1[b,p,n] = bb1[p] + sum_g v1[p,g]*x[b,g,n]   (and e2)
// ---------------------------------------------------------------------------
__global__ __launch_bounds__(256) void compute_e(
    const float* __restrict__ x,   // (B, G, N)
    const float* __restrict__ v1, const float* __restrict__ v2,
    const float* __restrict__ bb,
    float* __restrict__ e1, float* __restrict__ e2)
{
    int idx = blockIdx.x * blockDim.x + threadIdx.x;   // B*P*N threads
    int b = idx / (PP * NN);
    int rem = idx % (PP * NN);
    int p = rem / NN;
    int n = rem % NN;
    const float* xb = x + (long)b * GG * NN + n;
    const float* w1 = v1 + p * GG;
    const float* w2 = v2 + p * GG;
    float a1 = bb[p], a2 = bb[PP + p];
    for (int g = 0; g < GG; ++g) {
        float xv = xb[(long)g * NN];   // coalesced across n
        a1 += w1[g] * xv;
        a2 += w2[g] * xv;
    }
    e1[idx] = a1;
    e2[idx] = a2;
}

// ---------------------------------------------------------------------------
// Kernel 3: masked leaky-relu softmax row:  aij[b,p,i,:] (one block per row)
// ---------------------------------------------------------------------------
__global__ __launch_bounds__(256) void attn_softmax(
    const float* __restrict__ e1, const float* __restrict__ e2,
    const float* __restrict__ S,      // (B, 1, N, N)
    float* __restrict__ aij)          // (B*P, N, N)
{
    __shared__ float red[256];
    int rowId = blockIdx.x;                  // 0 .. B*P*N-1
    int b   = rowId / (PP * NN);
    int rem = rowId % (PP * NN);
    int p   = rem / NN;
    int i   = rem % NN;
    int tid = threadIdx.x;

    const float* e1row = e1 + (long)(b * PP + p) * NN;
    float e2i = e2[(long)(b * PP + p) * NN + i];
    const float* Srow = S + ((long)b * NN + i) * NN;

    float vals[4];
    bool  m[4];
    float lmax = -INFINITY;
#pragma unroll
    for (int t = 0; t < 4; ++t) {
        int j = tid + t * 256;
        float v = e1row[j] + e2i;
        v = v > 0.f ? v : ATT_SLOPE * v;               // leaky_relu(., 0.2)
        bool mk = fabsf(Srow[j]) > 1e-9f;
        vals[t] = mk ? v : -1.0e12f;                   // eij*mask - (1-mask)*INF
        m[t] = mk;
        lmax = fmaxf(lmax, vals[t]);
    }
    red[tid] = lmax; __syncthreads();
    for (int s = 128; s > 0; s >>= 1) {
        if (tid < s) red[tid] = fmaxf(red[tid], red[tid + s]);
        __syncthreads();
    }
    float rmax = red[0]; __syncthreads();

    float lsum = 0.f;
#pragma unroll
    for (int t = 0; t < 4; ++t) {
        float ex = m[t] ? expf(vals[t] - rmax) : 0.f;
        vals[t] = ex;
        lsum += ex;
    }
    red[tid] = lsum; __syncthreads();
    for (int s = 128; s > 0; s >>= 1) {
        if (tid < s) red[tid] += red[tid + s];
        __syncthreads();
    }
    float inv = 1.0f / red[0];                          // diag always present -> sum > 0
    float* arow = aij + (long)rowId * NN;
#pragma unroll
    for (int t = 0; t < 4; ++t) arow[tid + t * 256] = vals[t] * inv;
}

// ---------------------------------------------------------------------------
// Kernel 4: z[b,p,0,g,n] = x[b,g,n]
// ---------------------------------------------------------------------------
__global__ __launch_bounds__(256) void copy_x_to_z(
    const float* __restrict__ x, float* __restrict__ Z)
{
    long idx = (long)blockIdx.x * 256 + threadIdx.x;    // B*P*G*N
    int n = (int)(idx % NN); long r = idx / NN;
    int g = (int)(r % GG);   r /= GG;
    int p = (int)(r % PP);
    int b = (int)(r / PP);
    Z[(((long)(b * PP + p) * KCH + 0) * GG + g) * NN + n] = x[((long)b * GG + g) * NN + n];
}

// ---------------------------------------------------------------------------
// Async global -> LDS copy of one 16-byte chunk (gfx1250 ASYNCcnt path).
// LDS offset = low 32 bits of the generic pointer (ISA 10.2: LDS aperture
// truncates addr[31:0]).
// ---------------------------------------------------------------------------
__device__ __forceinline__ void async_copy_b128(void* lds_dst, const float* gsrc)
{
    unsigned lds_addr = (unsigned)(unsigned long long)lds_dst;
    asm volatile("global_load_async_to_lds_b128 %0, %1, off"
                 :: "v"(lds_addr), "v"(gsrc) : "memory");
}
__device__ __forceinline__ void wait_async_all()
{
    asm volatile("s_wait_asynccnt 0x0" ::: "memory");
}

// ---------------------------------------------------------------------------
// Batched f32 WMMA GEMM:  C[batch] = A[batch % aMod] @ B[batch]
//   A: row-major M x Kt,  B: row-major Kt x Nt,  C: row-major M x Nt
//   Block = 256 threads (8 waves) computes a 128x64 C tile; the 64-wide B
//   panel (KBLK x 64) is staged once per block into LDS with async
//   global->LDS B128 copies, double-buffered so the DMA overlaps the WMMA
//   loop (all 8 waves share it -> 8x less B global traffic).
//   V_WMMA_F32_16X16X4_F32 fragment layout (ISA 7.12.2):
//     A 16x4: lanes 0-15 hold K={0,1}, lanes 16-31 hold K={2,3}
//     B 4x16: VGPR0 = rows K{0,2}, VGPR1 = rows K{1,3}, lane -> column
//     C 16x16: VGPR r -> row r (lanes 0-15) / row r+8 (lanes 16-31)
//   Optional epilogue: +bias[row] then leaky_relu(0.01).
// ---------------------------------------------------------------------------
__global__ __launch_bounds__(256) void gemm_f32_wmma(
    const float* __restrict__ Abase, long aStride, int aMod,
    const float* __restrict__ Bbase, long bStride,
    float* __restrict__ Cbase, long cStride,
    int M, int Kt, int Nt,
    const float* __restrict__ bias)
{
    __shared__ float btile[2][KBLK][64];   // 16 KB double-buffered B panel

    int batch = blockIdx.y;
    const float* A  = Abase + (long)(batch % aMod) * aStride;
    const float* Bm = Bbase + (long)batch * bStride;
    float* C        = Cbase + (long)batch * cStride;

    int lane = threadIdx.x & 31;
    int wave = threadIdx.x >> 5;
    int colBlocks = Nt / 64;
    int rowBlock  = blockIdx.x / colBlocks;
    int colBlock  = blockIdx.x % colBlocks;
    int rowTile   = rowBlock * 8 + wave;            // 16-row tile per wave
    bool active   = rowTile * 16 < M;               // wave-uniform

    int row   = rowTile * 16 + (lane & 15);
    int colL  = lane & 15;                          // column within 64-wide panel
    int khalf = (lane >> 4) * 2;                    // 0 / 2 per ISA A,B striping
    const float* arow = A + (long)row * Kt;
    const float* gB   = Bm + colBlock * 64;         // B panel base (col offset)

    int tid = threadIdx.x;
    // ---- prologue: stage panel 0 into buffer 0 ----
#pragma unroll
    for (int c = 0; c < 2; ++c) {
        int chunk = tid + c * 256;                  // 512 x 16B = 32x64 floats
        int rrow  = chunk >> 4;
        int ccol  = (chunk & 15) << 2;
        async_copy_b128(&btile[0][rrow][ccol], gB + (long)rrow * Nt + ccol);
    }
    wait_async_all();
    __syncthreads();

    v8f acc0 = {}, acc1 = {}, acc2 = {}, acc3 = {};
    int buf = 0;
    for (int kk = 0; kk < Kt; kk += KBLK) {
        // ---- async-prefetch next panel into the other buffer ----
        if (kk + KBLK < Kt) {
            const float* gnext = gB + (long)(kk + KBLK) * Nt;
#pragma unroll
            for (int c = 0; c < 2; ++c) {
                int chunk = tid + c * 256;
                int rrow  = chunk >> 4;
                int ccol  = (chunk & 15) << 2;
                async_copy_b128(&btile[buf ^ 1][rrow][ccol],
                                gnext + (long)rrow * Nt + ccol);
            }
        }
        // ---- WMMA over the resident panel ----
        if (active) {
#pragma unroll
            for (int k4 = 0; k4 < KBLK; k4 += 4) {
                int kb = k4 + khalf;
                v2f a;
                a.x = arow[kk + kb];
                a.y = arow[kk + kb + 1];
                v2f b0, b1, b2, b3;
                b0.x = btile[buf][kb][colL];      b0.y = btile[buf][kb + 1][colL];
                b1.x = btile[buf][kb][colL + 16]; b1.y = btile[buf][kb + 1][colL + 16];
                b2.x = btile[buf][kb][colL + 32]; b2.y = btile[buf][kb + 1][colL + 32];
                b3.x = btile[buf][kb][colL + 48]; b3.y = btile[buf][kb + 1][colL + 48];
                acc0 = __builtin_amdgcn_wmma_f32_16x16x4_f32(false, a, false, b0, (short)0, acc0, false, false);
                acc1 = __builtin_amdgcn_wmma_f32_16x16x4_f32(false, a, false, b1, (short)0, acc1, false, false);
                acc2 = __builtin_amdgcn_wmma_f32_16x16x4_f32(false, a, false, b2, (short)0, acc2, false, false);
                acc3 = __builtin_amdgcn_wmma_f32_16x16x4_f32(false, a, false, b3, (short)0, acc3, false, false);
            }
        }
        wait_async_all();        // next panel landed in LDS
        __syncthreads();         // and prior panel reads are done block-wide
        buf ^= 1;
    }

    if (!active) return;
    int rbase = rowTile * 16 + 8 * (lane >> 4);
    int cbase = colBlock * 64 + (lane & 15);
#pragma unroll
    for (int r = 0; r < 8; ++r) {
        int rr = rbase + r;
        float o0 = acc0[r], o1 = acc1[r], o2 = acc2[r], o3 = acc3[r];
        if (bias) {
            float bv = bias[rr];
            o0 += bv; o1 += bv; o2 += bv; o3 += bv;
            o0 = o0 > 0.f ? o0 : OUT_SLOPE * o0;
            o1 = o1 > 0.f ? o1 : OUT_SLOPE * o1;
            o2 = o2 > 0.f ? o2 : OUT_SLOPE * o2;
            o3 = o3 > 0.f ? o3 : OUT_SLOPE * o3;
        }
        long base = (long)rr * Nt + cbase;
        C[base]      = o0;
        C[base + 16] = o1;
        C[base + 32] = o2;
        C[base + 48] = o3;
    }
}

// ---------------------------------------------------------------------------
extern "C" void kernel_launch(void* const* d_in, const int* in_sizes, int n_in,
                              void* d_out, int out_size, void* d_ws, size_t ws_size,
                              hipStream_t stream) {
    const float* x      = (const float*)d_in[0];   // (B,G,N)
    const float* mixer  = (const float*)d_in[1];   // (P,1,2F)
    const float* weight = (const float*)d_in[2];   // (P,1,F,G)
    const float* wbias  = (const float*)d_in[3];   // (P,1,F)
    const float* fw     = (const float*)d_in[4];   // (P,F,1,K,G) == (P, F, K*G)
    const float* bias   = (const float*)d_in[5];   // (F,1)
    const float* S      = (const float*)d_in[6];   // (B,1,N,N)
    float* out = (float*)d_out;                    // (B, P*F, N)
    float* ws  = (float*)d_ws;

    // workspace carve-out (floats)
    float* v1  = ws;                               // P*G
    float* v2  = ws + 1024;                        // P*G
    float* bb  = ws + 2048;                        // 2*P
    float* e1  = ws + 4096;                        // B*P*N
    float* e2  = e1 + (long)BB * PP * NN;          // B*P*N
    float* aij = ws + (1l << 17);                  // B*P*N*N  (128 MB)
    float* Z   = aij + (long)BB * PP * NN * NN;    // B*P*K*G*N (128 MB)

    precompute_v<<<dim3((PP * GG + 255) / 256), dim3(256), 0, stream>>>(
        mixer, weight, wbias, v1, v2, bb);

    compute_e<<<dim3((BB * PP * NN) / 256), dim3(256), 0, stream>>>(
        x, v1, v2, bb, e1, e2);

    attn_softmax<<<dim3(BB * PP * NN), dim3(256), 0, stream>>>(e1, e2, S, aij);

    copy_x_to_z<<<dim3((BB * PP * GG * NN) / 256), dim3(256), 0, stream>>>(x, Z);

    const long zb = (long)KCH * GG * NN;           // per-(b,p) z slab
    // grid.x = (M/128) * (Nt/64) = 2*16 = 32 blocks per batch
    dim3 gGrid(32, BB * PP);

    // attention chain: z_k = z_{k-1} @ aij   (M=G=256, Kt=N=1024, Nt=N=1024)
    for (int k = 1; k < KCH; ++k) {
        gemm_f32_wmma<<<gGrid, dim3(256), 0, stream>>>(
            Z + (long)(k - 1) * GG * NN, zb, BB * PP,
            aij, (long)NN * NN,
            Z + (long)k * GG * NN, zb,
            GG, NN, NN, nullptr);
    }

    // filter contraction + bias + leaky_relu(0.01) into final (B, P*F, N)
    gemm_f32_wmma<<<gGrid, dim3(256), 0, stream>>>(
        fw, (long)FF * KCH * GG, PP,               // A indexed by p = batch % P
        Z, zb,
        out, (long)FF * NN,
        FF, KCH * GG, NN, bias);
}